// Net_71889162600935
// MI455X (gfx1250) — compile-verified
//
#include <hip/hip_runtime.h>

typedef float v2f __attribute__((ext_vector_type(2)));
typedef float v8f __attribute__((ext_vector_type(8)));

#define BATCH 4
#define NPTS  4096
#define KNN   20
#define PEDGE (NPTS * KNN)   // 81920 edge columns per batch

#define MODE_PLAIN 0
#define MODE_EDGE  1
#define MODE_CAT3  2
#define MODE_CAT4  3

// ---------------------------------------------------------------------------
// x0 = concat(xyz, feat) zero-padded to 8 channels -> (B,8,N)
// ---------------------------------------------------------------------------
__global__ void k_build_x0(const float* __restrict__ xyz,
                           const float* __restrict__ feat,
                           float* __restrict__ x0) {
  int i = blockIdx.x * blockDim.x + threadIdx.x;
  int tot = BATCH * 8 * NPTS;
  if (i >= tot) return;
  int n = i % NPTS;
  int c = (i / NPTS) % 8;
  int b = i / (8 * NPTS);
  float v = 0.f;
  if (c < 3)      v = xyz[((size_t)b * 3 + c) * NPTS + n];
  else if (c < 6) v = feat[((size_t)b * 3 + (c - 3)) * NPTS + n];
  x0[i] = v;
}

// ---------------------------------------------------------------------------
// W9 (13x256) zero-padded to (64x256)
// ---------------------------------------------------------------------------
__global__ void k_pad_w9(const float* __restrict__ W9, float* __restrict__ Wp) {
  int i = blockIdx.x * blockDim.x + threadIdx.x;
  if (i >= 64 * 256) return;
  int o = i / 256;
  Wp[i] = (o < 13) ? W9[i] : 0.f;
}

// ---------------------------------------------------------------------------
// xx[b][n] = sum_c x[b][c][n]^2   (C = allocated channel count; pads are 0)
// ---------------------------------------------------------------------------
__global__ void k_rowsq(const float* __restrict__ x, int C,
                        float* __restrict__ xx) {
  int i = blockIdx.x * blockDim.x + threadIdx.x;
  if (i >= BATCH * NPTS) return;
  int n = i % NPTS;
  int b = i / NPTS;
  const float* p = x + (size_t)b * C * NPTS + n;
  float s = 0.f;
  for (int c = 0; c < C; ++c) {
    float v = p[(size_t)c * NPTS];
    s += v * v;
  }
  xx[i] = s;
}

// ---------------------------------------------------------------------------
// dist[m][n] = 2 * <x_m, x_n> - xx[m] - xx[n]   (one batch)
// One wave computes a 64x16 tile: 4 accumulators share each B fragment.
// Guard-free: C is a compile-time multiple of 4 (channels zero-padded).
// ---------------------------------------------------------------------------
template <int C>
__global__ void k_gram_dist(const float* __restrict__ x,
                            const float* __restrict__ xx,
                            int b, float* __restrict__ dist) {
  const int lane = threadIdx.x;
  const int lh = lane >> 4, lm = lane & 15;
  const int n0 = blockIdx.x * 16;
  const int m0 = blockIdx.y * 64;
  const float* xb = x + (size_t)b * C * NPTS;

  v8f acc[4] = {};
#pragma unroll 4
  for (int k0 = 0; k0 < C; k0 += 4) {
    v2f bf;
    bf.x = xb[(size_t)(k0 + lh) * NPTS + n0 + lm];
    bf.y = xb[(size_t)(k0 + 2 + lh) * NPTS + n0 + lm];
#pragma unroll
    for (int t = 0; t < 4; ++t) {
      v2f a;
      const int m = m0 + 16 * t + lm;
      a.x = xb[(size_t)(k0 + 2 * lh) * NPTS + m];
      a.y = xb[(size_t)(k0 + 1 + 2 * lh) * NPTS + m];
      acc[t] = __builtin_amdgcn_wmma_f32_16x16x4_f32(false, a, false, bf,
                                                     (short)0, acc[t],
                                                     false, false);
    }
  }

  const float xxn = xx[(size_t)b * NPTS + n0 + lm];
#pragma unroll
  for (int t = 0; t < 4; ++t) {
#pragma unroll
    for (int r = 0; r < 8; ++r) {
      int m = m0 + 16 * t + r + 8 * lh;
      dist[(size_t)m * NPTS + n0 + lm] =
          2.f * acc[t][r] - xx[(size_t)b * NPTS + m] - xxn;
    }
  }
}

// ---------------------------------------------------------------------------
// Top-20 per row of dist (one wave per row; dist is L2-resident, 64MB<192MB).
// float4 scan; selection order matches lax.top_k (value desc, index asc).
// ---------------------------------------------------------------------------
__global__ void k_topk(const float* __restrict__ dist, int b,
                       int* __restrict__ idx) {
  const int lane = threadIdx.x;
  const int n = blockIdx.x * blockDim.y + threadIdx.y;
  const float4* row4 = reinterpret_cast<const float4*>(dist + (size_t)n * NPTS);

  float pv = 3.402823466e38f;
  int pi = -1;
  for (int t = 0; t < KNN; ++t) {
    float best = -3.402823466e38f;
    int bidx = NPTS;
    for (int jb = lane; jb < NPTS / 4; jb += 32) {
      float4 v4 = row4[jb];
      int j = jb * 4;
      float vv[4] = {v4.x, v4.y, v4.z, v4.w};
#pragma unroll
      for (int u = 0; u < 4; ++u) {
        float v = vv[u];
        int jj = j + u;
        bool elig = (v < pv) || (v == pv && jj > pi);
        if (elig && (v > best || (v == best && jj < bidx))) { best = v; bidx = jj; }
      }
    }
#pragma unroll
    for (int off = 16; off > 0; off >>= 1) {
      float ov = __shfl_xor(best, off, 32);
      int oi = __shfl_xor(bidx, off, 32);
      if (ov > best || (ov == best && oi < bidx)) { best = ov; bidx = oi; }
    }
    idx[((size_t)b * NPTS + n) * KNN + t] = bidx;
    pv = best;
    pi = bidx;
  }
}

// ---------------------------------------------------------------------------
// Branch-free source ADDRESS selection (lowers to v_cndmask, not exec masks).
// CH = logical per-half channels (edge mode), CHA = allocated channel stride.
// ---------------------------------------------------------------------------
template <int MODE, int CIN, int CH, int CHA>
__device__ __forceinline__ const float*
src_addr(int b, int c, int col, int n, const float* s0, const float* s1,
         const float* s2, const float* s3, int gidx, int P) {
  if (MODE == MODE_PLAIN) {
    return s0 + ((size_t)b * CIN + c) * P + col;
  } else if (MODE == MODE_EDGE) {
    int cc = (c < CH) ? c : (c - CH);
    int pt = (c < CH) ? gidx : n;
    return s0 + ((size_t)b * CHA + cc) * NPTS + pt;
  } else if (MODE == MODE_CAT3) {
    const float* s = (c < 64) ? s0 : ((c < 128) ? s1 : s2);
    int cc = (c < 64) ? c : ((c < 128) ? (c - 64) : (c - 128));
    return s + ((size_t)b * 64 + cc) * NPTS + col;
  } else {  // MODE_CAT4: [0,1024) global feature (broadcast), then x1,x2,x3
    if (c < 1024) return s0 + (size_t)b * 1024 + c;
    const float* s = (c < 1088) ? s1 : ((c < 1152) ? s2 : s3);
    int cc = (c < 1088) ? (c - 1024) : ((c < 1152) ? (c - 1088) : (c - 1152));
    return s + ((size_t)b * 64 + cc) * NPTS + col;
  }
}

// ---------------------------------------------------------------------------
// Channel GEMM: out[b][o][col] = sum_c W[o][c] * src(b,c,col)
// One wave -> 64(O) x 16(col) tile; B fragment shared by 4 WMMAs.
// O must be a multiple of 64 (W9 is zero-padded); CIN a multiple of 4.
// ---------------------------------------------------------------------------
template <int MODE, int CIN, int CH, int CHA, bool GUARD>
__global__ void k_gemm(const float* __restrict__ W, int O, int P,
                       const float* __restrict__ s0, const float* __restrict__ s1,
                       const float* __restrict__ s2, const float* __restrict__ s3,
                       const int* __restrict__ idxbuf, float* __restrict__ out) {
  const int lane = threadIdx.x;
  const int lh = lane >> 4, lm = lane & 15;
  const int col0 = blockIdx.x * 16;
  const int m0 = blockIdx.y * 64;
  const int b = blockIdx.z;
  const int col = col0 + lm;

  int n = col, gidx = 0;
  if (MODE == MODE_EDGE) {
    n = col / KNN;
    int kk = col - n * KNN;
    gidx = idxbuf[((size_t)b * NPTS + n) * KNN + kk];
  }

  v8f acc[4] = {};
#pragma unroll 4
  for (int k0 = 0; k0 < CIN; k0 += 4) {
    v2f bf;
    bf.x = *src_addr<MODE, CIN, CH, CHA>(b, k0 + lh, col, n, s0, s1, s2, s3, gidx, P);
    bf.y = *src_addr<MODE, CIN, CH, CHA>(b, k0 + 2 + lh, col, n, s0, s1, s2, s3, gidx, P);
#pragma unroll
    for (int t = 0; t < 4; ++t) {
      v2f a;
      const int m = m0 + 16 * t + lm;
      a.x = W[(size_t)m * CIN + k0 + 2 * lh];
      a.y = W[(size_t)m * CIN + k0 + 1 + 2 * lh];
      acc[t] = __builtin_amdgcn_wmma_f32_16x16x4_f32(false, a, false, bf,
                                                     (short)0, acc[t],
                                                     false, false);
    }
  }

#pragma unroll
  for (int t = 0; t < 4; ++t) {
#pragma unroll
    for (int r = 0; r < 8; ++r) {
      int o = m0 + 16 * t + r + 8 * lh;
      if (!GUARD || o < O)
        out[((size_t)b * O + o) * P + col0 + lm] = acc[t][r];
    }
  }
}

// ---------------------------------------------------------------------------
// Per-channel sum / sumsq over (B, P): one block per channel, no atomics.
// ---------------------------------------------------------------------------
__global__ void k_bnstats(const float* __restrict__ y, int O, int P,
                          float* __restrict__ stats) {
  const int c = blockIdx.x;
  __shared__ float ssum[256], ssq[256];
  float s = 0.f, q = 0.f;
  for (int b = 0; b < BATCH; ++b) {
    const float4* p = reinterpret_cast<const float4*>(y + ((size_t)b * O + c) * P);
    for (int i = threadIdx.x; i < P / 4; i += 256) {
      float4 v = p[i];
      s += v.x + v.y + v.z + v.w;
      q += v.x * v.x + v.y * v.y + v.z * v.z + v.w * v.w;
    }
  }
  ssum[threadIdx.x] = s;
  ssq[threadIdx.x] = q;
  __syncthreads();
  for (int off = 128; off > 0; off >>= 1) {
    if (threadIdx.x < off) {
      ssum[threadIdx.x] += ssum[threadIdx.x + off];
      ssq[threadIdx.x] += ssq[threadIdx.x + off];
    }
    __syncthreads();
  }
  if (threadIdx.x == 0) {
    stats[2 * c] = ssum[0];
    stats[2 * c + 1] = ssq[0];
  }
}

// ---------------------------------------------------------------------------
// In-place BN (population var) + LeakyReLU(0.2), float4 per thread
// ---------------------------------------------------------------------------
__global__ void k_bn_lrelu(float* __restrict__ y,
                           const float* __restrict__ stats,
                           const float* __restrict__ g,
                           const float* __restrict__ bb, int O, int P) {
  size_t i4 = (size_t)blockIdx.x * blockDim.x + threadIdx.x;
  size_t tot4 = (size_t)BATCH * O * P / 4;
  if (i4 >= tot4) return;
  int c = (int)((i4 * 4 / P) % O);   // P % 4 == 0: group stays in one channel
  float cnt = (float)BATCH * (float)P;
  float mean = stats[2 * c] / cnt;
  float var = stats[2 * c + 1] / cnt - mean * mean;
  float sc = rsqrtf(var + 1e-5f) * g[c];
  float sh = bb[c];
  float4 v = reinterpret_cast<float4*>(y)[i4];
  float o[4] = {v.x, v.y, v.z, v.w};
#pragma unroll
  for (int u = 0; u < 4; ++u) {
    float t = (o[u] - mean) * sc + sh;
    o[u] = (t >= 0.f) ? t : 0.2f * t;
  }
  reinterpret_cast<float4*>(y)[i4] = make_float4(o[0], o[1], o[2], o[3]);
}

// ---------------------------------------------------------------------------
// max over the k (=20) neighbor dimension: (B,64,N,20) -> (B,64,N)
// ---------------------------------------------------------------------------
__global__ void k_maxk(const float* __restrict__ y, float* __restrict__ xo) {
  int i = blockIdx.x * blockDim.x + threadIdx.x;
  int tot = BATCH * 64 * NPTS;
  if (i >= tot) return;
  const float4* p = reinterpret_cast<const float4*>(y + (size_t)i * KNN);
  float m = -3.402823466e38f;
#pragma unroll
  for (int t = 0; t < KNN / 4; ++t) {
    float4 v = p[t];
    m = fmaxf(m, fmaxf(fmaxf(v.x, v.y), fmaxf(v.z, v.w)));
  }
  xo[i] = m;
}

// ---------------------------------------------------------------------------
// max over N: (B,O,N) -> (B,O); one block per (b,c)
// ---------------------------------------------------------------------------
__global__ void k_maxn(const float* __restrict__ y, float* __restrict__ xg) {
  int bc = blockIdx.x;
  __shared__ float sm[256];
  const float4* p = reinterpret_cast<const float4*>(y + (size_t)bc * NPTS);
  float m = -3.402823466e38f;
  for (int i = threadIdx.x; i < NPTS / 4; i += 256) {
    float4 v = p[i];
    m = fmaxf(m, fmaxf(fmaxf(v.x, v.y), fmaxf(v.z, v.w)));
  }
  sm[threadIdx.x] = m;
  __syncthreads();
  for (int off = 128; off > 0; off >>= 1) {
    if (threadIdx.x < off)
      sm[threadIdx.x] = fmaxf(sm[threadIdx.x], sm[threadIdx.x + off]);
    __syncthreads();
  }
  if (threadIdx.x == 0) xg[bc] = sm[0];
}

// ---------------------------------------------------------------------------
extern "C" void kernel_launch(void* const* d_in, const int* in_sizes, int n_in,
                              void* d_out, int out_size, void* d_ws,
                              size_t ws_size, hipStream_t stream) {
  const float* xyz  = (const float*)d_in[0];
  const float* feat = (const float*)d_in[1];
  const float* W1 = (const float*)d_in[2];
  const float* W2 = (const float*)d_in[3];
  const float* W3 = (const float*)d_in[4];
  const float* W4 = (const float*)d_in[5];
  const float* W5 = (const float*)d_in[6];
  const float* W6 = (const float*)d_in[7];
  const float* W7 = (const float*)d_in[8];
  const float* W8 = (const float*)d_in[9];
  const float* W9 = (const float*)d_in[10];
  const float* g[9];
  const float* bb[9];
  for (int i = 1; i <= 8; ++i) {
    g[i]  = (const float*)d_in[11 + 2 * (i - 1)];
    bb[i] = (const float*)d_in[12 + 2 * (i - 1)];
  }

  float* ws = (float*)d_ws;
  size_t off = 0;
  auto alloc = [&](size_t nf) { float* p = ws + off; off += nf; return p; };
  float* x0   = alloc((size_t)BATCH * 8 * NPTS);       // zero-padded to 8 ch
  float* xx   = alloc((size_t)BATCH * NPTS);
  float* dist = alloc((size_t)NPTS * NPTS);            // per-batch, reused
  int*   idx  = (int*)alloc((size_t)BATCH * NPTS * KNN);
  float* e1   = alloc((size_t)BATCH * 64 * PEDGE);
  float* e2   = alloc((size_t)BATCH * 64 * PEDGE);
  float* x1b  = alloc((size_t)BATCH * 64 * NPTS);
  float* x2b  = alloc((size_t)BATCH * 64 * NPTS);
  float* x3b  = alloc((size_t)BATCH * 64 * NPTS);
  float* y6   = alloc((size_t)BATCH * 1024 * NPTS);
  float* xg   = alloc((size_t)BATCH * 1024);
  float* y7   = alloc((size_t)BATCH * 512 * NPTS);
  float* y8   = alloc((size_t)BATCH * 256 * NPTS);
  float* Wp   = alloc((size_t)64 * 256);               // W9 zero-padded
  float* stats = alloc(2 * 1024);

  const dim3 wv(32);  // one wave (wave32)

  auto bn = [&](float* y, int O, int P, int li) {
    k_bnstats<<<dim3(O), 256, 0, stream>>>(y, O, P, stats);
    size_t tot4 = (size_t)BATCH * O * P / 4;
    k_bn_lrelu<<<(unsigned)((tot4 + 255) / 256), 256, 0, stream>>>(
        y, stats, g[li], bb[li], O, P);
  };

  // ---- stage 0: padded input concat + kNN on 8-padded 6-dim points
  k_build_x0<<<(BATCH * 8 * NPTS + 255) / 256, 256, 0, stream>>>(xyz, feat, x0);
  k_pad_w9<<<(64 * 256 + 255) / 256, 256, 0, stream>>>(W9, Wp);
  k_rowsq<<<(BATCH * NPTS + 255) / 256, 256, 0, stream>>>(x0, 8, xx);
  for (int b = 0; b < BATCH; ++b) {
    k_gram_dist<8><<<dim3(NPTS / 16, NPTS / 64), wv, 0, stream>>>(x0, xx, b, dist);
    k_topk<<<dim3(NPTS / 4), dim3(32, 4), 0, stream>>>(dist, b, idx);
  }

  // ---- edge block 1: W1 (64x12 edge) -> BN -> W2 (64x64) -> BN -> max_k
  k_gemm<MODE_EDGE, 12, 6, 8, false><<<dim3(PEDGE / 16, 1, BATCH), wv, 0, stream>>>(
      W1, 64, PEDGE, x0, nullptr, nullptr, nullptr, idx, e1);
  bn(e1, 64, PEDGE, 1);
  k_gemm<MODE_PLAIN, 64, 0, 0, false><<<dim3(PEDGE / 16, 1, BATCH), wv, 0, stream>>>(
      W2, 64, PEDGE, e1, nullptr, nullptr, nullptr, nullptr, e2);
  bn(e2, 64, PEDGE, 2);
  k_maxk<<<(BATCH * 64 * NPTS + 255) / 256, 256, 0, stream>>>(e2, x1b);

  // ---- edge block 2
  k_rowsq<<<(BATCH * NPTS + 255) / 256, 256, 0, stream>>>(x1b, 64, xx);
  for (int b = 0; b < BATCH; ++b) {
    k_gram_dist<64><<<dim3(NPTS / 16, NPTS / 64), wv, 0, stream>>>(x1b, xx, b, dist);
    k_topk<<<dim3(NPTS / 4), dim3(32, 4), 0, stream>>>(dist, b, idx);
  }
  k_gemm<MODE_EDGE, 128, 64, 64, false><<<dim3(PEDGE / 16, 1, BATCH), wv, 0, stream>>>(
      W3, 64, PEDGE, x1b, nullptr, nullptr, nullptr, idx, e1);
  bn(e1, 64, PEDGE, 3);
  k_gemm<MODE_PLAIN, 64, 0, 0, false><<<dim3(PEDGE / 16, 1, BATCH), wv, 0, stream>>>(
      W4, 64, PEDGE, e1, nullptr, nullptr, nullptr, nullptr, e2);
  bn(e2, 64, PEDGE, 4);
  k_maxk<<<(BATCH * 64 * NPTS + 255) / 256, 256, 0, stream>>>(e2, x2b);

  // ---- edge block 3
  k_rowsq<<<(BATCH * NPTS + 255) / 256, 256, 0, stream>>>(x2b, 64, xx);
  for (int b = 0; b < BATCH; ++b) {
    k_gram_dist<64><<<dim3(NPTS / 16, NPTS / 64), wv, 0, stream>>>(x2b, xx, b, dist);
    k_topk<<<dim3(NPTS / 4), dim3(32, 4), 0, stream>>>(dist, b, idx);
  }
  k_gemm<MODE_EDGE, 128, 64, 64, false><<<dim3(PEDGE / 16, 1, BATCH), wv, 0, stream>>>(
      W5, 64, PEDGE, x2b, nullptr, nullptr, nullptr, idx, e1);
  bn(e1, 64, PEDGE, 5);
  k_maxk<<<(BATCH * 64 * NPTS + 255) / 256, 256, 0, stream>>>(e1, x3b);

  // ---- W6: (1024 x 192) on concat(x1,x2,x3), BN, global max over N
  k_gemm<MODE_CAT3, 192, 0, 0, false><<<dim3(NPTS / 16, 1024 / 64, BATCH), wv, 0, stream>>>(
      W6, 1024, NPTS, x1b, x2b, x3b, nullptr, nullptr, y6);
  bn(y6, 1024, NPTS, 6);
  k_maxn<<<dim3(BATCH * 1024), 256, 0, stream>>>(y6, xg);

  // ---- W7: (512 x 1216) on concat(global, x1, x2, x3)
  k_gemm<MODE_CAT4, 1216, 0, 0, false><<<dim3(NPTS / 16, 512 / 64, BATCH), wv, 0, stream>>>(
      W7, 512, NPTS, xg, x1b, x2b, x3b, nullptr, y7);
  bn(y7, 512, NPTS, 7);

  // ---- W8: (256 x 512)
  k_gemm<MODE_PLAIN, 512, 0, 0, false><<<dim3(NPTS / 16, 256 / 64, BATCH), wv, 0, stream>>>(
      W8, 256, NPTS, y7, nullptr, nullptr, nullptr, nullptr, y8);
  bn(y8, 256, NPTS, 8);

  // ---- W9: (13 x 256) via zero-padded 64x256 weights, guarded store
  k_gemm<MODE_PLAIN, 256, 0, 0, true><<<dim3(NPTS / 16, 1, BATCH), wv, 0, stream>>>(
      Wp, 13, NPTS, y8, nullptr, nullptr, nullptr, nullptr, (float*)d_out);
}